// PCN_28681791603104
// MI455X (gfx1250) — compile-verified
//
#include <hip/hip_runtime.h>

typedef float v2f __attribute__((ext_vector_type(2)));
typedef float v8f __attribute__((ext_vector_type(8)));

#define LR      0.01f
#define CYCLES  10
#define NL      14          // number of error layers E[0..13]
#define DIM     64
#define MAT     4096        // 64*64
#define THREADS 512         // 16 wave32
#define NJOBS   (NL * 16)   // 14 layers * (4x4) 16x16 output tiles = 224

static __device__ __forceinline__ v8f wmma_f32_k4(v2f a, v2f b, v8f c) {
#if defined(__gfx1250__)
    // V_WMMA_F32_16X16X4_F32 : D(16x16 f32) = A(16x4 f32) * B(4x16 f32) + C
    return __builtin_amdgcn_wmma_f32_16x16x4_f32(
        /*neg_a=*/false, a, /*neg_b=*/false, b,
        /*c_mod=*/(short)0, c, /*reuse_a=*/false, /*reuse_b=*/false);
#else
    (void)a; (void)b;
    return c; // host pass / non-gfx1250: never executed
#endif
}

extern "C" __global__ __launch_bounds__(THREADS)
void pcn_persistent_kernel(const float* __restrict__ S_in,  // (15,64,64)
                           const float* __restrict__ Wk,    // (14,64,64) row-major (out,in)
                           const float* __restrict__ Wg_in, // (64,64)
                           float* __restrict__ out,         // 8192: [S_fin[-1] | Wg_fin]
                           float* __restrict__ Sws)         // scratch: 15*4096 floats
{
    extern __shared__ float smem[];
    float* Elds   = smem;             // 14*4096 floats = 224 KB
    float* wgbuf0 = smem + NL * MAT;  // 4096 floats
    float* wgbuf1 = wgbuf0 + MAT;     // 4096 floats   (total LDS = 256 KB)

    const int tid  = threadIdx.x;
    const int lane = tid & 31;
    const int wave = tid >> 5;
    const int nwav = THREADS >> 5;
    const int lx   = lane & 15;   // N / M position within tile
    const int lh   = lane >> 4;   // half-wave selects K pair (0 -> K{0,1}, 1 -> K{2,3})

    // ---- init: stage S into scratch, Wg into LDS buffer 0 (inputs untouched) ----
    for (int i = tid; i < 15 * MAT; i += THREADS) Sws[i] = S_in[i];
    for (int i = tid; i < MAT;      i += THREADS) wgbuf0[i] = Wg_in[i];
    __threadfence();
    __syncthreads();

    int cur = 0;
    for (int cyc = 0; cyc < CYCLES; ++cyc) {
        // ================= phase 1: E[l] = S[l] - relu(S[l+1]) @ Wk[l]^T =================
        for (int job = wave; job < NJOBS; job += nwav) {
            const int l  = job >> 4;
            const int m0 = ((job >> 2) & 3) * 16;
            const int n0 = (job & 3) * 16;
            const float* A  = Sws + (size_t)(l + 1) * MAT; // (b,i) row-major, relu applied on load
            const float* Bk = Wk  + (size_t)l * MAT;       // (o,i) row-major -> B[k][n] = Wk[n][k]
            v8f acc = {};
            for (int k0 = 0; k0 < DIM; k0 += 4) {
                const int ka = k0 + lh * 2;
                const float* ap = A  + (m0 + lx) * DIM + ka;
                const float* bp = Bk + (n0 + lx) * DIM + ka;
                v2f a, b;
                a.x = fmaxf(ap[0], 0.0f); a.y = fmaxf(ap[1], 0.0f);
                b.x = bp[0];              b.y = bp[1];
                acc = wmma_f32_k4(a, b, acc);
            }
            const float* Sl = Sws  + (size_t)l * MAT;
            float*       El = Elds + (size_t)l * MAT;
#pragma unroll
            for (int r = 0; r < 8; ++r) {
                const int idx = (m0 + r + 8 * lh) * DIM + n0 + lx; // C/D VGPR layout
                El[idx] = Sl[idx] - acc[r];
            }
        }
        __syncthreads();

        // ====== phase 2: G[l] = Wg^T @ E_next[l]; S update; Wg replacement ======
        const float* wgcur  = cur ? wgbuf1 : wgbuf0;
        float*       wgnext = cur ? wgbuf0 : wgbuf1;
        for (int job = wave; job < NJOBS; job += nwav) {
            const int l  = job >> 4;
            const int m0 = ((job >> 2) & 3) * 16;
            const int n0 = (job & 3) * 16;
            const int ln = (l < NL - 1) ? l + 1 : NL - 1;  // E_next
            const float* Bn = Elds + (size_t)ln * MAT;     // B[k][n] = E_next[k][n]
            v8f acc = {};
            for (int k0 = 0; k0 < DIM; k0 += 4) {
                const int ka = k0 + lh * 2;
                v2f a, b;
                a.x = wgcur[ka * DIM + m0 + lx];        // A[m][k] = Wg[k][m]
                a.y = wgcur[(ka + 1) * DIM + m0 + lx];
                b.x = Bn[ka * DIM + n0 + lx];
                b.y = Bn[(ka + 1) * DIM + n0 + lx];
                acc = wmma_f32_k4(a, b, acc);
            }
            float*       Sl = Sws  + (size_t)l * MAT;
            const float* El = Elds + (size_t)l * MAT;
#pragma unroll
            for (int r = 0; r < 8; ++r) {
                const int idx = (m0 + r + 8 * lh) * DIM + n0 + lx;
                const float s = Sl[idx];
                const float e = El[idx];
                Sl[idx] = s + LR * (-e + (s > 0.0f ? 1.0f : 0.0f)) * acc[r];
            }
        }
        // Wg_new[i][j] = LR * E[13][i][j] * relu'(S[14])[j][i]   (full replacement)
        {
            const float* E13 = Elds + (size_t)(NL - 1) * MAT;
            const float* S14 = Sws  + (size_t)14 * MAT;          // constant across cycles
            for (int idx = tid; idx < MAT; idx += THREADS) {
                const int i = idx >> 6, j = idx & 63;
                wgnext[idx] = LR * E13[idx] * (S14[j * DIM + i] > 0.0f ? 1.0f : 0.0f);
            }
        }
        cur ^= 1;
        __threadfence();   // make S (global scratch) stores visible before next cycle
        __syncthreads();
    }

    // ---- outputs: S_fin[-1] == initial S[14]; Wg_fin from active LDS buffer ----
    const float* wgfin = cur ? wgbuf1 : wgbuf0;
    for (int i = tid; i < MAT; i += THREADS) {
        out[i]       = S_in[14 * MAT + i];
        out[MAT + i] = wgfin[i];
    }
}

extern "C" void kernel_launch(void* const* d_in, const int* in_sizes, int n_in,
                              void* d_out, int out_size, void* d_ws, size_t ws_size,
                              hipStream_t stream) {
    // inputs: 0 input_data (unused), 1 s0 (unused), 2 S (15,64,64),
    //         3 W1 (unused), 4 Wk (14,64,64), 5 Wg (64,64)
    const float* S  = (const float*)d_in[2];
    const float* Wk = (const float*)d_in[4];
    const float* Wg = (const float*)d_in[5];
    float* out = (float*)d_out;
    float* Sws = (float*)d_ws;   // needs 15*4096*4 = 240 KB

    const size_t shmem = (size_t)(NL * MAT + 2 * MAT) * sizeof(float); // 256 KB LDS
    (void)hipFuncSetAttribute((const void*)pcn_persistent_kernel,
                              hipFuncAttributeMaxDynamicSharedMemorySize, (int)shmem);

    pcn_persistent_kernel<<<dim3(1), dim3(THREADS), shmem, stream>>>(S, Wk, Wg, out, Sws);
}